// model_0_74182675137054
// MI455X (gfx1250) — compile-verified
//
#include <hip/hip_runtime.h>
#include <hip/hip_bf16.h>

// ---------------------------------------------------------------------------
// Llama-style 4-layer transformer forward for MI455X (gfx1250), fp32.
// GEMMs via V_WMMA_F32_16X16X4_F32 fed from double-buffered LDS filled with
// GLOBAL_LOAD_ASYNC_TO_LDS_B128 (ASYNCcnt DMA, no VGPR round trip).
// Workload is HBM-bound (~850MB fp32 weights vs 424 GFLOP), so fp32 WMMA +
// async data movement is the right operating point.
// ---------------------------------------------------------------------------

#define DIM   2048
#define NH    16
#define HD    128
#define HID   5632
#define NL    4
#define SEQ   512
#define BSZ   2
#define MROWS (BSZ * SEQ)   // 1024

#define KB    32            // K chunk staged per double-buffer step
#define LDP   36            // LDS row stride (floats): 144B, 16B-aligned,
                            // 36*r mod 64 conflict-free over 16 rows

typedef float v8f __attribute__((ext_vector_type(8)));
typedef float v2f __attribute__((ext_vector_type(2)));

// Async DMA: global -> LDS, 16 bytes per lane, tracked by ASYNCcnt.
__device__ __forceinline__ void async_ld_b128(unsigned lds_off,
                                              const float* g) {
  asm volatile("global_load_async_to_lds_b128 %0, %1, off"
               :: "v"(lds_off), "v"(g)
               : "memory");
}

// ---------------------------------------------------------------------------
// Embedding gather: h[m, c] = tok_emb[tokens[m], c]
// ---------------------------------------------------------------------------
__global__ __launch_bounds__(256) void embed_kernel(
    const int* __restrict__ tokens, const float* __restrict__ tok_emb,
    float* __restrict__ h) {
  size_t idx = (size_t)blockIdx.x * 256 + threadIdx.x;  // over MROWS*DIM
  int m = (int)(idx / DIM);
  int c = (int)(idx % DIM);
  int tok = tokens[m];
  h[idx] = tok_emb[(size_t)tok * DIM + c];
}

// ---------------------------------------------------------------------------
// RMSNorm: one block (256 threads) per row of DIM=2048.
// ---------------------------------------------------------------------------
__global__ __launch_bounds__(256) void rmsnorm_kernel(
    const float* __restrict__ x, const float* __restrict__ w,
    float* __restrict__ y) {
  __shared__ float red[256];
  const int row = blockIdx.x;
  const int t = threadIdx.x;
  const float* xr = x + (size_t)row * DIM;
  float ss = 0.f;
#pragma unroll
  for (int i = 0; i < DIM / 256; ++i) {
    float v = xr[t + i * 256];
    ss += v * v;
  }
  red[t] = ss;
  __syncthreads();
  for (int s = 128; s > 0; s >>= 1) {
    if (t < s) red[t] += red[t + s];
    __syncthreads();
  }
  float r = rsqrtf(red[0] * (1.0f / DIM) + 1e-6f);
  float* yr = y + (size_t)row * DIM;
#pragma unroll
  for (int i = 0; i < DIM / 256; ++i) {
    int c = t + i * 256;
    yr[c] = xr[c] * r * w[c];
  }
}

// ---------------------------------------------------------------------------
// WMMA fp32 GEMM: Y[M,N] = X[M,K] @ W[N,K]^T  (+ optional residual)
// WG = 256 threads (8 waves); WG tile 64(M) x 128(N); wave tile 32x32.
// K staged in KB=32 chunks through double-buffered LDS using async-to-LDS
// DMA; each chunk consumed as 8 steps of V_WMMA_F32_16X16X4_F32 (x4 tiles).
//
// fp32 WMMA fragment layout (ISA 7.12.2):
//   A 16x4 : lanes 0-15 -> M=lane, V0=K0,V1=K1 ; lanes 16-31 -> V0=K2,V1=K3
//   B 4x16 : lanes 0-15 -> N=lane, V0=K0,V1=K1 ; lanes 16-31 -> V0=K2,V1=K3
//   C 16x16: VGPR r -> M=r (lanes 0-15), M=8+r (lanes 16-31); N=lane&15
// ---------------------------------------------------------------------------
__global__ __launch_bounds__(256) void gemm_wmma_kernel(
    const float* __restrict__ X, const float* __restrict__ W,
    const float* __restrict__ res, float* __restrict__ Y,
    int M, int N, int K) {
  __shared__ float Xs[2][64][LDP];
  __shared__ float Ws[2][128][LDP];

  const int tid = threadIdx.x;
  const int mBase = blockIdx.y * 64;
  const int nBase = blockIdx.x * 128;
  const int wid = tid >> 5;
  const int lane = tid & 31;
  const int half = lane >> 4;      // 0: K0..1 / M=r ; 1: K2..3 / M=8+r
  const int l16 = lane & 15;
  const int waveM = wid & 1;       // 2 waves along M
  const int waveN = wid >> 1;      // 4 waves along N

  v8f c[2][2] = {};                // 4 accum tiles of 16x16 fp32

  // Per-thread staging coordinates (one 64x32 X tile + one 128x32 W tile).
  const int xRow = tid >> 2;            // 0..63
  const int xCol = (tid & 3) << 3;      // 0,8,16,24   (2x b128)
  const int wRow = tid >> 1;            // 0..127
  const int wCol = (tid & 1) << 4;      // 0,16        (4x b128)

  const float* Xg = X + (size_t)(mBase + xRow) * K + xCol;
  const float* Wg = W + (size_t)(nBase + wRow) * K + wCol;

  // LDS byte offsets (low 32 bits of the flat shared address == LDS offset).
  unsigned xL[2], wL[2];
  xL[0] = (unsigned)(size_t)&Xs[0][xRow][xCol];
  xL[1] = (unsigned)(size_t)&Xs[1][xRow][xCol];
  wL[0] = (unsigned)(size_t)&Ws[0][wRow][wCol];
  wL[1] = (unsigned)(size_t)&Ws[1][wRow][wCol];

  // Prologue: chunk 0 -> buffer 0 (6 async b128 per thread).
  async_ld_b128(xL[0] +  0, Xg + 0);
  async_ld_b128(xL[0] + 16, Xg + 4);
  async_ld_b128(wL[0] +  0, Wg + 0);
  async_ld_b128(wL[0] + 16, Wg + 4);
  async_ld_b128(wL[0] + 32, Wg + 8);
  async_ld_b128(wL[0] + 48, Wg + 12);

  const int nChunks = K / KB;
  for (int ci = 0; ci < nChunks; ++ci) {
    const int cur = ci & 1;
    __syncthreads();  // all waves done reading buf[cur^1] (prev compute)
    if (ci + 1 < nChunks) {
      const int nb = cur ^ 1;
      const float* Xn = Xg + (size_t)(ci + 1) * KB;
      const float* Wn = Wg + (size_t)(ci + 1) * KB;
      async_ld_b128(xL[nb] +  0, Xn + 0);
      async_ld_b128(xL[nb] + 16, Xn + 4);
      async_ld_b128(wL[nb] +  0, Wn + 0);
      async_ld_b128(wL[nb] + 16, Wn + 4);
      async_ld_b128(wL[nb] + 32, Wn + 8);
      async_ld_b128(wL[nb] + 48, Wn + 12);
      // 6 newer ops in flight; async ops retire in order, so <=6 means the
      // 6 loads of chunk `ci` have landed in LDS.
      asm volatile("s_wait_asynccnt 6" ::: "memory");
    } else {
      asm volatile("s_wait_asynccnt 0" ::: "memory");
    }
    __syncthreads();  // publish chunk `ci` across the workgroup

#pragma unroll
    for (int ks = 0; ks < KB; ks += 4) {
      const int ka = ks + half * 2;
      v2f a[2], b[2];
#pragma unroll
      for (int i = 0; i < 2; ++i) {
        const int arow = waveM * 32 + i * 16 + l16;
        a[i].x = Xs[cur][arow][ka];
        a[i].y = Xs[cur][arow][ka + 1];
        const int bcol = waveN * 32 + i * 16 + l16;
        b[i].x = Ws[cur][bcol][ka];
        b[i].y = Ws[cur][bcol][ka + 1];
      }
#pragma unroll
      for (int i = 0; i < 2; ++i)
#pragma unroll
        for (int j = 0; j < 2; ++j)
          c[i][j] = __builtin_amdgcn_wmma_f32_16x16x4_f32(
              false, a[i], false, b[j], (short)0, c[i][j], false, false);
    }
  }

  // Store (+ optional residual)
#pragma unroll
  for (int i = 0; i < 2; ++i)
#pragma unroll
    for (int j = 0; j < 2; ++j) {
      const int n = nBase + waveN * 32 + j * 16 + l16;
#pragma unroll
      for (int r = 0; r < 8; ++r) {
        const int mrow = mBase + waveM * 32 + i * 16 + half * 8 + r;
        const size_t idx = (size_t)mrow * N + n;
        float v = c[i][j][r];
        if (res) v += res[idx];
        Y[idx] = v;
      }
    }
}

// ---------------------------------------------------------------------------
// RoPE in place on (MROWS, NH, HD) packed as [m, h*HD + d]; start_pos = 0.
// One thread per (m, h, i) pair, i in [0, HD/2).
// ---------------------------------------------------------------------------
__global__ __launch_bounds__(256) void rope_kernel(float* __restrict__ x) {
  const int idx = blockIdx.x * 256 + threadIdx.x;   // MROWS*NH*64
  const int i = idx & 63;
  const int h = (idx >> 6) & (NH - 1);
  const int m = idx >> 10;
  const int pos = m & (SEQ - 1);
  // freq = THETA^{-(2i/HD)} = exp(-i * ln(1e4)/64)
  const float freq = __expf(-(float)i * (9.210340371976184f / 64.0f));
  const float ang = (float)pos * freq;
  const float cs = __cosf(ang);
  const float sn = __sinf(ang);
  float* p = x + (size_t)m * DIM + h * HD + 2 * i;
  const float xr = p[0];
  const float xi = p[1];
  p[0] = xr * cs - xi * sn;
  p[1] = xr * sn + xi * cs;
}

// ---------------------------------------------------------------------------
// Causal attention, start_pos = 0: one wave per query row, online softmax.
// Each lane holds 4 dims (float4) of q / v-accumulator; dot reduced by shfl.
// ---------------------------------------------------------------------------
__global__ __launch_bounds__(256) void attn_kernel(
    const float* __restrict__ Q, const float* __restrict__ K,
    const float* __restrict__ V, float* __restrict__ O) {
  const int lane = threadIdx.x & 31;
  const int w = blockIdx.x * 8 + (threadIdx.x >> 5);  // BSZ*NH*SEQ waves
  const int q = w & (SEQ - 1);
  const int h = (w >> 9) & (NH - 1);
  const int b = w >> 13;

  const size_t rowQ = (size_t)(b * SEQ + q) * DIM + h * HD + lane * 4;
  const float4 qv = *(const float4*)(Q + rowQ);

  float4 acc = make_float4(0.f, 0.f, 0.f, 0.f);
  float mx = -3.0e38f, l = 0.f;
  const float inv = 0.08838834764831845f;  // 1/sqrt(128)

  for (int kk = 0; kk <= q; ++kk) {
    const size_t rowK = (size_t)(b * SEQ + kk) * DIM + h * HD + lane * 4;
    const float4 kv = *(const float4*)(K + rowK);
    float p = qv.x * kv.x + qv.y * kv.y + qv.z * kv.z + qv.w * kv.w;
#pragma unroll
    for (int off = 16; off; off >>= 1) p += __shfl_xor(p, off, 32);
    const float s = p * inv;
    const float mn = fmaxf(mx, s);
    const float scale = __expf(mx - mn);
    const float pe = __expf(s - mn);
    l = l * scale + pe;
    const float4 vv = *(const float4*)(V + rowK);
    acc.x = acc.x * scale + pe * vv.x;
    acc.y = acc.y * scale + pe * vv.y;
    acc.z = acc.z * scale + pe * vv.z;
    acc.w = acc.w * scale + pe * vv.w;
    mx = mn;
  }
  const float rl = 1.0f / l;
  float4 o = make_float4(acc.x * rl, acc.y * rl, acc.z * rl, acc.w * rl);
  *(float4*)(O + rowQ) = o;
}

// ---------------------------------------------------------------------------
// h1 = silu(h1) * h3   (elementwise over MROWS*HID)
// ---------------------------------------------------------------------------
__global__ __launch_bounds__(256) void silu_mul_kernel(
    float* __restrict__ h1, const float* __restrict__ h3) {
  const size_t i = (size_t)blockIdx.x * 256 + threadIdx.x;
  const float x = h1[i];
  const float sig = 1.0f / (1.0f + __expf(-x));
  h1[i] = x * sig * h3[i];
}

// ---------------------------------------------------------------------------
extern "C" void kernel_launch(void* const* d_in, const int* in_sizes, int n_in,
                              void* d_out, int out_size, void* d_ws,
                              size_t ws_size, hipStream_t stream) {
  const int*   tokens  = (const int*)d_in[0];
  // d_in[1] = start_pos (== 0, unused)
  const float* tok_emb = (const float*)d_in[2];
  const float* wq      = (const float*)d_in[3];
  const float* wk      = (const float*)d_in[4];
  const float* wv      = (const float*)d_in[5];
  const float* wo      = (const float*)d_in[6];
  const float* w1      = (const float*)d_in[7];
  const float* w2      = (const float*)d_in[8];
  const float* w3      = (const float*)d_in[9];
  const float* anw     = (const float*)d_in[10];
  const float* fnw     = (const float*)d_in[11];

  float* h = (float*)d_out;  // (BSZ, SEQ, DIM) lives in the output buffer

  const size_t SZ_D = (size_t)MROWS * DIM;   // 2,097,152 floats
  const size_t SZ_H = (size_t)MROWS * HID;   // 5,767,168 floats
  float* xn   = (float*)d_ws;
  float* qb   = xn + SZ_D;
  float* kb   = qb + SZ_D;
  float* vb   = kb + SZ_D;
  float* atb  = vb + SZ_D;
  float* h1b  = atb + SZ_D;
  float* h3b  = h1b + SZ_H;

  const dim3 blk(256);

  // Embedding
  embed_kernel<<<dim3((unsigned)(SZ_D / 256)), blk, 0, stream>>>(tokens,
                                                                 tok_emb, h);

  const dim3 gSq(DIM / 128, MROWS / 64);   // N=2048 GEMMs
  const dim3 gUp(HID / 128, MROWS / 64);   // N=5632 GEMMs
  const dim3 gNorm(MROWS);
  const unsigned ropeBlocks = (unsigned)((size_t)MROWS * NH * (HD / 2) / 256);
  const unsigned attnBlocks = (unsigned)(BSZ * NH * SEQ / 8);
  const unsigned siluBlocks = (unsigned)(SZ_H / 256);

  for (int l = 0; l < NL; ++l) {
    const float* lwq = wq + (size_t)l * DIM * DIM;
    const float* lwk = wk + (size_t)l * DIM * DIM;
    const float* lwv = wv + (size_t)l * DIM * DIM;
    const float* lwo = wo + (size_t)l * DIM * DIM;
    const float* lw1 = w1 + (size_t)l * HID * DIM;
    const float* lw2 = w2 + (size_t)l * DIM * HID;
    const float* lw3 = w3 + (size_t)l * HID * DIM;
    const float* lan = anw + (size_t)l * DIM;
    const float* lfn = fnw + (size_t)l * DIM;

    // --- attention block ---
    rmsnorm_kernel<<<gNorm, blk, 0, stream>>>(h, lan, xn);
    gemm_wmma_kernel<<<gSq, blk, 0, stream>>>(xn, lwq, nullptr, qb,
                                              MROWS, DIM, DIM);
    gemm_wmma_kernel<<<gSq, blk, 0, stream>>>(xn, lwk, nullptr, kb,
                                              MROWS, DIM, DIM);
    gemm_wmma_kernel<<<gSq, blk, 0, stream>>>(xn, lwv, nullptr, vb,
                                              MROWS, DIM, DIM);
    rope_kernel<<<ropeBlocks, blk, 0, stream>>>(qb);
    rope_kernel<<<ropeBlocks, blk, 0, stream>>>(kb);
    attn_kernel<<<attnBlocks, blk, 0, stream>>>(qb, kb, vb, atb);
    gemm_wmma_kernel<<<gSq, blk, 0, stream>>>(atb, lwo, h, h,
                                              MROWS, DIM, DIM);

    // --- FFN block ---
    rmsnorm_kernel<<<gNorm, blk, 0, stream>>>(h, lfn, xn);
    gemm_wmma_kernel<<<gUp, blk, 0, stream>>>(xn, lw1, nullptr, h1b,
                                              MROWS, HID, DIM);
    gemm_wmma_kernel<<<gUp, blk, 0, stream>>>(xn, lw3, nullptr, h3b,
                                              MROWS, HID, DIM);
    silu_mul_kernel<<<siluBlocks, blk, 0, stream>>>(h1b, h3b);
    gemm_wmma_kernel<<<gSq, blk, 0, stream>>>(h1b, lw2, h, h,
                                              MROWS, DIM, HID);
  }
}